// BOW_model_84954453115688
// MI455X (gfx1250) — compile-verified
//
#include <hip/hip_runtime.h>
#include <math.h>

#define VOCAB        100000
#define HIDDEN       512
#define BATCH        4096
#define TOTAL_TOKENS 819200
#define BN_EPS       1e-5f

typedef __attribute__((ext_vector_type(2))) float v2f;
typedef __attribute__((ext_vector_type(8))) float v8f;

// ---------------------------------------------------------------------------
// Kernel 1: segment mean-pool.  segment_ids are sorted, so block b binary-
// searches its [start,end) token range, then streams embedding rows with
// fully coalesced loads (thread t owns columns t and t+256) and prefetches
// the next token's row (global_prefetch_b8).
// ---------------------------------------------------------------------------
__global__ void __launch_bounds__(256)
pool_kernel(const int* __restrict__ tokens,
            const int* __restrict__ seg,
            const float* __restrict__ emb,
            float* __restrict__ pooled) {
    const int b   = blockIdx.x;
    const int tid = threadIdx.x;

    // lower_bound(seg, b)
    int lo = 0, hi = TOTAL_TOKENS;
    while (lo < hi) { int mid = (lo + hi) >> 1; if (seg[mid] < b) lo = mid + 1; else hi = mid; }
    const int start = lo;
    // lower_bound(seg, b+1)
    hi = TOTAL_TOKENS;
    while (lo < hi) { int mid = (lo + hi) >> 1; if (seg[mid] < b + 1) lo = mid + 1; else hi = mid; }
    const int end = lo;

    const int c0 = tid, c1 = tid + 256;
    float acc0 = 0.0f, acc1 = 0.0f;
    for (int i = start; i < end; ++i) {
        const float* row = emb + (size_t)tokens[i] * HIDDEN;
        if (i + 1 < end) {
            const float* nrow = emb + (size_t)tokens[i + 1] * HIDDEN;
            __builtin_prefetch(nrow + c0, 0, 1);
            __builtin_prefetch(nrow + c1, 0, 1);
        }
        acc0 += row[c0];
        acc1 += row[c1];
    }
    const float inv = 1.0f / fmaxf((float)(end - start), 1.0f);
    pooled[(size_t)b * HIDDEN + c0] = acc0 * inv;
    pooled[(size_t)b * HIDDEN + c1] = acc1 * inv;
}

// ---------------------------------------------------------------------------
// Kernel 2: H = pooled @ W_h + b_h  via V_WMMA_F32_16X16X4_F32 (fp32 exact
// path; GEMM is compute-trivial vs the HBM-bound gather so fp32 is free).
// Each wave produces a 16(M) x 64(N) strip: 4 accumulators reuse one A frag.
//   A frag (16x4): lane l<16 -> {A[m][k],A[m][k+1]}, lane>=16 -> {A[m][k+2],A[m][k+3]}
//   B frag (4x16): lane l<16 -> {B[k][n],B[k+1][n]}, lane>=16 -> {B[k+2][n],B[k+3][n]}
//   D: VGPR r = row mBase + r + 8*(lane>>4), col nBase + (lane&15)
// ---------------------------------------------------------------------------
__global__ void __launch_bounds__(128)
gemm_kernel(const float* __restrict__ A,      // pooled [4096,512]
            const float* __restrict__ B,      // W_h    [512,512]
            const float* __restrict__ bias,   // b_h    [512]
            float* __restrict__ H) {          // h      [4096,512]
    const int tid  = threadIdx.x;
    const int wave = tid >> 5;
    const int lane = tid & 31;
    const int tile = blockIdx.x * 4 + wave;     // 0..2047
    const int tm   = tile >> 3;                 // 256 M-tiles
    const int tn   = tile & 7;                  // 8 N-groups of 64
    const int mBase = tm * 16;
    const int nBase = tn * 64;
    const int l    = lane & 15;
    const int half = lane >> 4;
    const int kOff = half * 2;

    v8f c0 = {}, c1 = {}, c2 = {}, c3 = {};
    const float* aRow = A + (size_t)(mBase + l) * HIDDEN + kOff;

    for (int k = 0; k < HIDDEN; k += 4) {
        v2f a = *(const v2f*)(aRow + k);
        const float* bRow = B + (size_t)(k + kOff) * HIDDEN + l;
        v2f b0 = { bRow[nBase +  0], bRow[HIDDEN + nBase +  0] };
        v2f b1 = { bRow[nBase + 16], bRow[HIDDEN + nBase + 16] };
        v2f b2 = { bRow[nBase + 32], bRow[HIDDEN + nBase + 32] };
        v2f b3 = { bRow[nBase + 48], bRow[HIDDEN + nBase + 48] };
        c0 = __builtin_amdgcn_wmma_f32_16x16x4_f32(false, a, false, b0, (short)0, c0, false, false);
        c1 = __builtin_amdgcn_wmma_f32_16x16x4_f32(false, a, false, b1, (short)0, c1, false, false);
        c2 = __builtin_amdgcn_wmma_f32_16x16x4_f32(false, a, false, b2, (short)0, c2, false, false);
        c3 = __builtin_amdgcn_wmma_f32_16x16x4_f32(false, a, false, b3, (short)0, c3, false, false);
    }

    #pragma unroll
    for (int r = 0; r < 8; ++r) {
        const int m = mBase + r + 8 * half;
        float* hRow = H + (size_t)m * HIDDEN + nBase + l;
        hRow[ 0] = c0[r] + bias[nBase + l +  0];
        hRow[16] = c1[r] + bias[nBase + l + 16];
        hRow[32] = c2[r] + bias[nBase + l + 32];
        hRow[48] = c3[r] + bias[nBase + l + 48];
    }
}

// ---------------------------------------------------------------------------
// Kernel 3: batch-norm statistics (training mode, biased variance).
// One thread per column; reads are coalesced across adjacent columns.
// Emits fused affine: ss[c]=gamma*rsqrt(var+eps), ss[512+c]=beta-mu*scale.
// ---------------------------------------------------------------------------
__global__ void __launch_bounds__(256)
bn_stats_kernel(const float* __restrict__ H,
                const float* __restrict__ gamma,
                const float* __restrict__ beta,
                float* __restrict__ ss) {
    const int col = blockIdx.x * 256 + threadIdx.x;
    float s = 0.0f, s2 = 0.0f;
    for (int r = 0; r < BATCH; ++r) {
        float v = H[(size_t)r * HIDDEN + col];
        s  += v;
        s2 += v * v;
    }
    const float mu  = s * (1.0f / BATCH);
    const float var = fmaxf(s2 * (1.0f / BATCH) - mu * mu, 0.0f);
    const float sc  = gamma[col] * rsqrtf(var + BN_EPS);
    ss[col]          = sc;
    ss[HIDDEN + col] = beta[col] - mu * sc;
}

// ---------------------------------------------------------------------------
// Kernel 4: apply BN+ReLU and reduce against W_o -> logits[row] (+b_o).
// ---------------------------------------------------------------------------
__global__ void __launch_bounds__(256)
logits_kernel(const float* __restrict__ H,
              const float* __restrict__ ss,
              const float* __restrict__ Wo,
              const float* __restrict__ bo,
              float* __restrict__ logits) {
    __shared__ float red[256];
    const int row = blockIdx.x;
    const int tid = threadIdx.x;
    const int c0 = tid, c1 = tid + 256;
    const float* hRow = H + (size_t)row * HIDDEN;
    float v0 = fmaxf(hRow[c0] * ss[c0] + ss[HIDDEN + c0], 0.0f);
    float v1 = fmaxf(hRow[c1] * ss[c1] + ss[HIDDEN + c1], 0.0f);
    red[tid] = v0 * Wo[c0] + v1 * Wo[c1];
    __syncthreads();
    for (int s = 128; s > 0; s >>= 1) {
        if (tid < s) red[tid] += red[tid + s];
        __syncthreads();
    }
    if (tid == 0) logits[row] = red[0] + bo[0];
}

// ---------------------------------------------------------------------------
// Kernel 5: loss = mean(softplus(x) - t*x) over 4096 logits.
// ---------------------------------------------------------------------------
__global__ void __launch_bounds__(256)
loss_kernel(const float* __restrict__ logits,
            const float* __restrict__ t,
            float* __restrict__ loss) {
    __shared__ float red[256];
    const int tid = threadIdx.x;
    float acc = 0.0f;
    for (int i = tid; i < BATCH; i += 256) {
        float x  = logits[i];
        float sp = fmaxf(x, 0.0f) + log1pf(expf(-fabsf(x)));   // stable softplus
        acc += sp - t[i] * x;
    }
    red[tid] = acc;
    __syncthreads();
    for (int s = 128; s > 0; s >>= 1) {
        if (tid < s) red[tid] += red[tid + s];
        __syncthreads();
    }
    if (tid == 0) loss[0] = red[0] * (1.0f / BATCH);
}

// ---------------------------------------------------------------------------
extern "C" void kernel_launch(void* const* d_in, const int* in_sizes, int n_in,
                              void* d_out, int out_size, void* d_ws, size_t ws_size,
                              hipStream_t stream) {
    const int*   tokens = (const int*)  d_in[0];
    const int*   seg    = (const int*)  d_in[1];
    const float* t      = (const float*)d_in[2];
    const float* emb    = (const float*)d_in[3];
    const float* Wh     = (const float*)d_in[4];
    const float* bh     = (const float*)d_in[5];
    const float* gamma  = (const float*)d_in[6];
    const float* beta   = (const float*)d_in[7];
    const float* Wo     = (const float*)d_in[8];
    const float* bo     = (const float*)d_in[9];

    float* out    = (float*)d_out;            // [0]=loss, [1..4096]=logits
    float* loss   = out;
    float* logits = out + 1;

    float* pooled = (float*)d_ws;                         // [4096,512]  8 MB
    float* H      = pooled + (size_t)BATCH * HIDDEN;      // [4096,512]  8 MB
    float* ss     = H      + (size_t)BATCH * HIDDEN;      // [2,512]     4 KB

    pool_kernel   <<<BATCH, 256, 0, stream>>>(tokens, seg, emb, pooled);
    gemm_kernel   <<<512, 128, 0, stream>>>(pooled, Wh, bh, H);   // 2048 waves = 256 M-tiles x 8 N-groups
    bn_stats_kernel<<<HIDDEN / 256, 256, 0, stream>>>(H, gamma, beta, ss);
    logits_kernel <<<BATCH, 256, 0, stream>>>(H, ss, Wo, bo, logits);
    loss_kernel   <<<1, 256, 0, stream>>>(logits, t, loss);
}